// LDAHead_49297634623814
// MI455X (gfx1250) — compile-verified
//
#include <hip/hip_runtime.h>
#include <hip/hip_bf16.h>
#include <math.h>

// Problem constants (from reference)
#define NB   4096   // batch
#define ND   128    // feature dim
#define NC   512    // classes
#define EPSF 1e-5f

typedef __attribute__((ext_vector_type(2))) float v2f;
typedef __attribute__((ext_vector_type(8))) float v8f;

// ---------------- workspace layout (floats) ----------------
#define OFF_COUNTS 0
#define OFF_MEAN   512
#define OFF_POOL   66048
#define OFF_P      82432
#define OFF_PMU    98816
#define OFF_R      164352
#define OFF_LP     164864
#define OFF_Q      165376
#define N_ZERO     82432   // counts + meansum + pooledacc are contiguous

// ---------------- kernel 1: zero accumulators ----------------
__global__ void k_zero(float* __restrict__ ws, int n) {
    int i = blockIdx.x * blockDim.x + threadIdx.x;
    if (i < n) ws[i] = 0.0f;
}

// ---------------- kernel 2: segment sums (counts, mean sums) ----------------
__global__ void k_accum(const float* __restrict__ z, const int* __restrict__ y,
                        float* __restrict__ counts, float* __restrict__ meansum) {
    int idx = blockIdx.x * blockDim.x + threadIdx.x;   // [0, NB*ND)
    int b = idx >> 7;
    int d = idx & (ND - 1);
    int cls = y[b];
    atomicAdd(&meansum[cls * ND + d], z[idx]);
    if (d == 0) atomicAdd(&counts[cls], 1.0f);
}

// ---------------- kernel 3: finalize mean, log prior ----------------
__global__ void k_finalize(float* __restrict__ counts, float* __restrict__ mean,
                           float* __restrict__ lp, float invTotal) {
    int idx = blockIdx.x * blockDim.x + threadIdx.x;   // [0, NC*ND)
    int c = idx >> 7;
    int d = idx & (ND - 1);
    float cnt = counts[c] + EPSF;
    mean[c * ND + d] /= cnt;
    if (d == 0) lp[c] = logf(cnt * invTotal);
}

// ---------------- kernel 4: pooled = sum_b dz dz^T (accumulator) ----------------
__global__ void k_pooled(const float* __restrict__ z, const int* __restrict__ y,
                         const float* __restrict__ mean, float* __restrict__ pool) {
    __shared__ float dz[ND];
    int t = threadIdx.x;
    int j = t & (ND - 1);
    int half = t >> 7;           // 0 or 1
    float acc[64];
#pragma unroll
    for (int k = 0; k < 64; ++k) acc[k] = 0.0f;

    int base = blockIdx.x * 256;
    for (int s = 0; s < 256; ++s) {
        int b = base + s;
        if (t < ND) {
            int cls = y[b];
            dz[t] = z[b * ND + t] - mean[cls * ND + t];
        }
        __syncthreads();
        float dzj = dz[j];
#pragma unroll
        for (int k = 0; k < 64; ++k) {
            acc[k] += dz[2 * k + half] * dzj;
        }
        __syncthreads();
    }
#pragma unroll
    for (int k = 0; k < 64; ++k) {
        int i = 2 * k + half;
        atomicAdd(&pool[i * ND + j], acc[k]);
    }
}

// ---------------- kernel 5: Cholesky + cholesky_inverse (1 block, 128 thr) ----
__global__ void k_solver(const float* __restrict__ poolacc, float* __restrict__ P,
                         float invTotal) {
    extern __shared__ float smem[];
    float* A  = smem;                 // working matrix / Cholesky factor (lower)
    float* Li = smem + ND * ND;       // L^{-1}
    int t = threadIdx.x;              // 0..127

    for (int j = 0; j < ND; ++j) {
        float v = 0.5f * (poolacc[t * ND + j] + poolacc[j * ND + t]) * invTotal;
        if (j == t) v += EPSF;
        A[t * ND + j] = v;
    }
    __syncthreads();

    for (int k = 0; k < ND; ++k) {
        if (t == k) A[k * ND + k] = sqrtf(A[k * ND + k]);
        __syncthreads();
        if (t > k) A[t * ND + k] /= A[k * ND + k];
        __syncthreads();
        if (t > k) {
            float lik = A[t * ND + k];
            for (int j = k + 1; j <= t; ++j)
                A[t * ND + j] -= lik * A[j * ND + k];
        }
        __syncthreads();
    }

    for (int i = t; i < ND; ++i) {
        float s = (i == t) ? 1.0f : 0.0f;
        for (int k = t; k < i; ++k) s -= A[i * ND + k] * Li[k * ND + t];
        Li[i * ND + t] = s / A[i * ND + i];
    }
    __syncthreads();

    for (int j = 0; j < ND; ++j) {
        int kmin = (t > j) ? t : j;
        float s = 0.0f;
        for (int k = kmin; k < ND; ++k) s += Li[k * ND + t] * Li[k * ND + j];
        P[t * ND + j] = s;
    }
}

// ---------------- kernel 6: Pmu = mean @ P, r_c = mu_c . Pmu_c --------------
__global__ void k_pmur(const float* __restrict__ mean, const float* __restrict__ P,
                       float* __restrict__ Pmu, float* __restrict__ r) {
    __shared__ float mu[ND];
    __shared__ float red[ND];
    int c = blockIdx.x;
    int t = threadIdx.x;              // 0..127
    mu[t] = mean[c * ND + t];
    __syncthreads();
    float s = 0.0f;
    for (int j = 0; j < ND; ++j) s += P[t * ND + j] * mu[j];   // P symmetric
    Pmu[c * ND + t] = s;
    red[t] = s * mu[t];
    __syncthreads();
    for (int off = 64; off > 0; off >>= 1) {
        if (t < off) red[t] += red[t + off];
        __syncthreads();
    }
    if (t == 0) r[c] = red[0];
}

// ---------------- kernel 7: q_b = z_b^T P z_b -------------------------------
__global__ void k_q(const float* __restrict__ z, const float* __restrict__ P,
                    float* __restrict__ q) {
    __shared__ float zr[ND];
    __shared__ float red[ND];
    int b = blockIdx.x;
    int t = threadIdx.x;
    zr[t] = z[b * ND + t];
    __syncthreads();
    float s = 0.0f;
    for (int j = 0; j < ND; ++j) s += P[t * ND + j] * zr[j];
    red[t] = s * zr[t];
    __syncthreads();
    for (int off = 64; off > 0; off >>= 1) {
        if (t < off) red[t] += red[t + off];
        __syncthreads();
    }
    if (t == 0) q[b] = red[0];
}

// ---------------- kernel 8: LDS-tiled WMMA GEMM + fused epilogue ------------
// cross = z @ Pmu^T ; out[b,c] = lp[c] - 0.5*(q[b] - 2*cross[b,c] + r[c])
// Block (256 thr, 8 waves): 32 rows x 256 cols of output.
//   - z tile   32x128  staged coalesced into LDS (rows padded to 130 floats)
//   - Pmu slab 256x128 staged coalesced into LDS (padded)
// Wave: 32x32 output = 2x2 WMMA tiles, 4 v8f accumulators.
// Per K-step: 4 conflict-free ds_load_b64 fragments feed 4 v_wmma_f32_16x16x4_f32.
#define LDP 130                              // padded LDS row (bank-conflict-free)
#define GEMM_LDS_FLOATS (32 * LDP + 256 * LDP)

__global__ void k_gemm(const float* __restrict__ z, const float* __restrict__ Pmu,
                       const float* __restrict__ q, const float* __restrict__ r,
                       const float* __restrict__ lp, float* __restrict__ out) {
    extern __shared__ float sm[];
    float* za = sm;               // [32][LDP]
    float* pb = sm + 32 * LDP;    // [256][LDP]

    int t  = threadIdx.x;
    int mb = blockIdx.x >> 1;     // 128 M-blocks of 32 rows
    int nb = blockIdx.x & 1;      // 2 N-blocks of 256 cols
    int m0 = mb * 32;
    int n0 = nb * 256;

    // stage A tile (coalesced global, linear LDS writes)
    for (int idx = t; idx < 32 * 128; idx += 256) {
        int row = idx >> 7, col = idx & 127;
        za[row * LDP + col] = z[(size_t)(m0 + row) * ND + col];
    }
    // stage B slab (coalesced)
    for (int idx = t; idx < 256 * 128; idx += 256) {
        int row = idx >> 7, col = idx & 127;
        pb[row * LDP + col] = Pmu[(size_t)(n0 + row) * ND + col];
    }
    __syncthreads();

    int w    = t >> 5;            // wave 0..7 -> 32-col strip
    int lane = t & 31;
    int half = lane >> 4;         // K sub-pair select
    int lr   = lane & 15;

    const float* arow0 = za + lr * LDP;              // A tile rows 0..15
    const float* arow1 = za + (16 + lr) * LDP;       // A tile rows 16..31
    const float* brow0 = pb + (w * 32 + lr) * LDP;   // B cols 0..15 of strip
    const float* brow1 = pb + (w * 32 + 16 + lr) * LDP;

    v8f acc00 = {}, acc01 = {}, acc10 = {}, acc11 = {};
#pragma unroll 8
    for (int kk = 0; kk < ND / 4; ++kk) {
        int ko = kk * 4 + 2 * half;
        v2f a0 = *(const v2f*)(arow0 + ko);
        v2f a1 = *(const v2f*)(arow1 + ko);
        v2f b0 = *(const v2f*)(brow0 + ko);
        v2f b1 = *(const v2f*)(brow1 + ko);
        acc00 = __builtin_amdgcn_wmma_f32_16x16x4_f32(false, a0, false, b0, (short)0, acc00, false, false);
        acc01 = __builtin_amdgcn_wmma_f32_16x16x4_f32(false, a0, false, b1, (short)0, acc01, false, false);
        acc10 = __builtin_amdgcn_wmma_f32_16x16x4_f32(false, a1, false, b0, (short)0, acc10, false, false);
        acc11 = __builtin_amdgcn_wmma_f32_16x16x4_f32(false, a1, false, b1, (short)0, acc11, false, false);
    }

    int c0g = n0 + w * 32 + lr;       // global col, tile col 0
    int c1g = c0g + 16;               // tile col 1
    float rc0 = r[c0g],  rc1 = r[c1g];
    float lp0 = lp[c0g], lp1 = lp[c1g];
#pragma unroll
    for (int v = 0; v < 8; ++v) {
        int b0r = m0 + v + 8 * half;  // C/D layout: VGPR v -> M=v (+8 hi half)
        int b1r = b0r + 16;
        float qb0 = q[b0r], qb1 = q[b1r];
        out[(size_t)b0r * NC + c0g] = lp0 - 0.5f * (qb0 - 2.0f * acc00[v] + rc0);
        out[(size_t)b0r * NC + c1g] = lp1 - 0.5f * (qb0 - 2.0f * acc01[v] + rc1);
        out[(size_t)b1r * NC + c0g] = lp0 - 0.5f * (qb1 - 2.0f * acc10[v] + rc0);
        out[(size_t)b1r * NC + c1g] = lp1 - 0.5f * (qb1 - 2.0f * acc11[v] + rc1);
    }
}

// ---------------- launcher ----------------
extern "C" void kernel_launch(void* const* d_in, const int* in_sizes, int n_in,
                              void* d_out, int out_size, void* d_ws, size_t ws_size,
                              hipStream_t stream) {
    (void)in_sizes; (void)n_in; (void)out_size; (void)ws_size;
    const float* z = (const float*)d_in[0];
    const int*   y = (const int*)d_in[1];
    float* ws  = (float*)d_ws;
    float* out = (float*)d_out;

    float* counts = ws + OFF_COUNTS;
    float* mean   = ws + OFF_MEAN;
    float* pool   = ws + OFF_POOL;
    float* P      = ws + OFF_P;
    float* Pmu    = ws + OFF_PMU;
    float* r      = ws + OFF_R;
    float* lp     = ws + OFF_LP;
    float* q      = ws + OFF_Q;

    const float invTotal = 1.0f / ((float)NB + (float)NC * EPSF);

    k_zero<<<(N_ZERO + 255) / 256, 256, 0, stream>>>(ws, N_ZERO);
    k_accum<<<(NB * ND) / 256, 256, 0, stream>>>(z, y, counts, mean);
    k_finalize<<<(NC * ND) / 256, 256, 0, stream>>>(counts, mean, lp, invTotal);
    k_pooled<<<NB / 256, 256, 0, stream>>>(z, y, mean, pool);
    k_solver<<<1, ND, 2 * ND * ND * sizeof(float), stream>>>(pool, P, invTotal);
    k_pmur<<<NC, ND, 0, stream>>>(mean, P, Pmu, r);
    k_q<<<NB, ND, 0, stream>>>(z, P, q);
    // 128 M-blocks x 2 N-blocks; ~146 KB dynamic LDS per block
    k_gemm<<<(NB / 32) * (NC / 256), 256, GEMM_LDS_FLOATS * sizeof(float), stream>>>(
        z, Pmu, q, r, lp, out);
}